// SindyAutoencoder_44805098832301
// MI455X (gfx1250) — compile-verified
//
#include <hip/hip_runtime.h>

// ---------------------------------------------------------------------------
// SINDy autoencoder forward + analytic JVPs on gfx1250 (wave32, WMMA f32).
//
// Strategy: never materialize Jacobians. Each layer does a fused
//   H   = relu(A @ W^T + b)
//   Vo  = (pre>0) * (Vin @ W^T)          (forward-mode JVP through ReLU MLP)
// as one wave-per-16x16-tile WMMA kernel using V_WMMA_F32_16X16X4_F32
// (fp32 tensor op -> matches reference precision; problem is BW-bound).
// The SINDy library GEMM builds theta on the fly from z (z cached in LDS).
// ---------------------------------------------------------------------------

typedef __attribute__((ext_vector_type(2))) float v2f;
typedef __attribute__((ext_vector_type(8))) float v8f;

#define WMMA_F32(a, b, c) \
  __builtin_amdgcn_wmma_f32_16x16x4_f32(false, (a), false, (b), (short)0, (c), false, false)

#define BATCH 4096
#define DLAT  64
#define NPAIR 2016          // 64*63/2
#define KLIB  2080          // DLAT + NPAIR (bias column folded into C init)

// ---------------------------------------------------------------------------
// Fused layer: H = relu(A @ W^T + bias); Vout = (pre>0) * (Vin @ W^T)
// A,Vin: [B,K] row-major. W: [N,K] (torch Linear layout). H,Vout: [B,N].
// One wave per 16x16 output tile; K must be a multiple of 4; N of 16.
// Grid is sized so (B/16)*(N/16) waves exist exactly (no divergent guards:
// WMMA requires EXEC all ones).
// ---------------------------------------------------------------------------
__global__ __launch_bounds__(256) void layer_jvp_kernel(
    const float* __restrict__ A, const float* __restrict__ Vin,
    const float* __restrict__ W, const float* __restrict__ bias,
    float* __restrict__ H, float* __restrict__ Vout, int K, int N)
{
  const int lane  = threadIdx.x & 31;
  const int wave  = (blockIdx.x * blockDim.x + threadIdx.x) >> 5;
  const int tilesN = N >> 4;
  const int tileM  = wave / tilesN;
  const int tileN  = wave % tilesN;

  const int mrow = lane & 15;
  const int row  = tileM * 16 + mrow;     // A/Vin row for this lane (A-matrix M)
  const int col  = tileN * 16 + mrow;     // output column == W row (B-matrix N)
  const int kz   = (lane >> 4) << 1;      // lane half selects K pair {0,1} / {2,3}

  const float* arow = A   + (size_t)row * K + kz;
  const float* vrow = Vin + (size_t)row * K + kz;
  const float* wrow = W   + (size_t)col * K + kz;   // B[k][n] = W[n][k] -> contiguous

  v8f acc  = {0.f, 0.f, 0.f, 0.f, 0.f, 0.f, 0.f, 0.f};
  v8f accv = {0.f, 0.f, 0.f, 0.f, 0.f, 0.f, 0.f, 0.f};

  #pragma unroll 4
  for (int k = 0; k < K; k += 4) {
    v2f a = *(const v2f*)(arow + k);
    v2f b = *(const v2f*)(wrow + k);
    v2f v = *(const v2f*)(vrow + k);
    acc  = WMMA_F32(a, b, acc);    // primal
    accv = WMMA_F32(v, b, accv);   // JVP (shares B operand)
  }

  // Epilogue: bias + ReLU on primal, mask the JVP with (pre > 0).
  const float bcol  = bias[col];
  const int   mbase = tileM * 16 + ((lane >> 4) << 3);  // C/D rows: VGPR r -> m = r + 8*(lane/16)
  #pragma unroll
  for (int r = 0; r < 8; ++r) {
    float pre = acc[r] + bcol;
    bool  on  = pre > 0.f;
    size_t o  = (size_t)(mbase + r) * N + col;
    H[o]    = on ? pre : 0.f;
    Vout[o] = on ? accv[r] : 0.f;
  }
}

// ---------------------------------------------------------------------------
// SINDy: out[b,n] = Xi[0,n] + sum_k z[b,k]*Xi[1+k,n] + sum_p z_i z_j * Xi[65+p,n]
// GEMM over the 2080 non-bias library columns with A built on the fly;
// bias row folded into the accumulator init. z tile (16x64) lives in LDS.
// ---------------------------------------------------------------------------
__global__ __launch_bounds__(256) void sindy_kernel(
    const float* __restrict__ z, const float* __restrict__ Xi,
    const unsigned short* __restrict__ tab, float* __restrict__ out)
{
  __shared__ float zs[8][16][DLAT];       // one 16x64 z tile per wave = 32 KB

  const int lane  = threadIdx.x & 31;
  const int wslot = threadIdx.x >> 5;
  const int wave  = (blockIdx.x * blockDim.x + threadIdx.x) >> 5;
  const int tileM = wave >> 2;            // tilesN = 64/16 = 4
  const int tileN = wave & 3;

  // Stage this wave's 16x64 z tile into LDS.
  for (int idx = lane; idx < 16 * DLAT; idx += 32)
    zs[wslot][idx >> 6][idx & 63] =
        z[(size_t)(tileM * 16 + (idx >> 6)) * DLAT + (idx & 63)];
  __syncthreads();

  const int mrow = lane & 15;
  const int col  = tileN * 16 + mrow;
  const int kz   = (lane >> 4) << 1;
  const float* zr = zs[wslot][mrow];

  v8f acc;
  const float bterm = Xi[col];            // theta[:,0] == 1  ->  + Xi[0, col]
  #pragma unroll
  for (int r = 0; r < 8; ++r) acc[r] = bterm;

  for (int k = 0; k < KLIB; k += 4) {
    v2f a, b;
    #pragma unroll
    for (int t = 0; t < 2; ++t) {
      int e = k + kz + t;                 // library column 1+e
      float th;
      if (e < DLAT) {
        th = zr[e];                       // linear term (wave-uniform branch)
      } else {
        unsigned ij = tab[e - DLAT];      // packed (i<<8)|j, i<j
        th = zr[ij >> 8] * zr[ij & 255];  // pairwise product
      }
      a[t] = th;
      b[t] = Xi[(size_t)(e + 1) * DLAT + col];
    }
    acc = WMMA_F32(a, b, acc);
  }

  const int mbase = tileM * 16 + ((lane >> 4) << 3);
  #pragma unroll
  for (int r = 0; r < 8; ++r)
    out[(size_t)(mbase + r) * DLAT + col] = acc[r];
}

// ---------------------------------------------------------------------------
// Prep: XiEff = XI * XI_mask; pair index table for triu(64, k=1), row-major.
// ---------------------------------------------------------------------------
__global__ void prep_kernel(const float* __restrict__ XI,
                            const float* __restrict__ XIm,
                            float* __restrict__ XiEff,
                            unsigned short* __restrict__ tab)
{
  int i = blockIdx.x * blockDim.x + threadIdx.x;
  if (i < (1 + DLAT + NPAIR) * DLAT) XiEff[i] = XI[i] * XIm[i];
  if (i < NPAIR) {
    int a = 0, c = 0;
    while (c + (DLAT - 1 - a) <= i) { c += DLAT - 1 - a; ++a; }
    int b = a + 1 + (i - c);
    tab[i] = (unsigned short)((a << 8) | b);
  }
}

// ---------------------------------------------------------------------------
extern "C" void kernel_launch(void* const* d_in, const int* in_sizes, int n_in,
                              void* d_out, int out_size, void* d_ws, size_t ws_size,
                              hipStream_t stream)
{
  const float* x    = (const float*)d_in[0];
  const float* xdot = (const float*)d_in[1];
  const float* We1  = (const float*)d_in[2];
  const float* be1  = (const float*)d_in[3];
  const float* We2  = (const float*)d_in[4];
  const float* be2  = (const float*)d_in[5];
  const float* We3  = (const float*)d_in[6];
  const float* be3  = (const float*)d_in[7];
  const float* Wd1  = (const float*)d_in[8];
  const float* bd1  = (const float*)d_in[9];
  const float* Wd2  = (const float*)d_in[10];
  const float* bd2  = (const float*)d_in[11];
  const float* Wd3  = (const float*)d_in[12];
  const float* bd3  = (const float*)d_in[13];
  const float* XI   = (const float*)d_in[14];
  const float* XIm  = (const float*)d_in[15];
  (void)in_sizes; (void)n_in; (void)out_size; (void)ws_size;

  // Output layout: (xh, dxh_dt, z, dz_dt, dz_dt_sindy) concatenated flat.
  float* out = (float*)d_out;
  float* xh  = out;
  float* dxh = xh  + (size_t)BATCH * 512;
  float* zo  = dxh + (size_t)BATCH * 512;
  float* dz  = zo  + (size_t)BATCH * DLAT;
  float* dzs = dz  + (size_t)BATCH * DLAT;

  // Workspace: XiEff | pair table | two [B,128] + two [B,64] reusable buffers.
  float*          ws    = (float*)d_ws;
  float*          XiEff = ws;                               // 2081*64 floats
  unsigned short* tab   = (unsigned short*)(ws + 133184);   // 2016 u16
  float* bufA = ws + 133184 + 1024;                         // [B,128]  h1 -> g2
  float* bufB = bufA + (size_t)BATCH * 128;                 // [B,128]  v1 -> u2
  float* bufC = bufB + (size_t)BATCH * 128;                 // [B,64]   h2 -> g1
  float* bufD = bufC + (size_t)BATCH * DLAT;                // [B,64]   v2 -> u1

  hipLaunchKernelGGL(prep_kernel, dim3((133184 + 255) / 256), dim3(256), 0, stream,
                     XI, XIm, XiEff, tab);

  auto layer = [&](const float* A, const float* V, const float* W, const float* b,
                   float* H, float* Vo, int K, int N) {
    int waves = (BATCH / 16) * (N / 16);   // always a multiple of 8 here
    hipLaunchKernelGGL(layer_jvp_kernel, dim3(waves / 8), dim3(256), 0, stream,
                       A, V, W, b, H, Vo, K, N);
  };

  // Encoder (primal + JVP with tangent xdot)
  layer(x,    xdot, We1, be1, bufA, bufB, 512, 128);   // h1, v1
  layer(bufA, bufB, We2, be2, bufC, bufD, 128, 64);    // h2, v2
  layer(bufC, bufD, We3, be3, zo,   dz,   64,  64);    // z,  dz_dt

  // SINDy library GEMM: (B/16)*(64/16) = 1024 waves = 128 blocks of 8 waves
  hipLaunchKernelGGL(sindy_kernel, dim3(128), dim3(256), 0, stream,
                     zo, XiEff, tab, dzs);

  // Decoder (primal + JVP with tangent dz_dt_sindy)
  layer(zo,   dzs,  Wd1, bd1, bufC, bufD, 64,  64);    // g1, u1
  layer(bufC, bufD, Wd2, bd2, bufA, bufB, 64,  128);   // g2, u2
  layer(bufA, bufB, Wd3, bd3, xh,   dxh,  128, 512);   // xh, dxh_dt
}